// Bert_AvgPooling_77704548319257
// MI455X (gfx1250) — compile-verified
//
#include <hip/hip_runtime.h>
#include <hip/hip_bf16.h>

// Problem constants (match reference: B=64, S=512, H=768, NC=1536)
#define HID        768
#define NTOK       (64 * 512)
#define NCLAUSE    1536
#define TOK_PER_BLK 64        // tokens walked per accumulation block
#define COLS_PER_THR 3        // 256 threads * 3 cols = 768

typedef float v2f __attribute__((ext_vector_type(2)));
typedef float v8f __attribute__((ext_vector_type(8)));

// ---------------------------------------------------------------------------
// Kernel 1: zero the [NC x H] f32 sums accumulator (4.7 MB in d_ws)
// ---------------------------------------------------------------------------
__global__ __launch_bounds__(256)
void zero_f4_kernel(float4* __restrict__ p, int n4) {
    int i = blockIdx.x * blockDim.x + threadIdx.x;
    int stride = gridDim.x * blockDim.x;
    float4 z = make_float4(0.f, 0.f, 0.f, 0.f);
    for (; i < n4; i += stride) p[i] = z;
}

// ---------------------------------------------------------------------------
// Kernel 2: segment-sum of token rows into sums[NC][H].
// Block = 256 threads, each thread owns 3 consecutive hidden columns
// (coalesced 12B = global_load_b96 per thread -> whole 3KB row per token).
// Walk TOK_PER_BLK consecutive tokens; keep a running register accumulator
// and flush with global_atomic_add_f32 only when the segment id changes.
// Correct for arbitrary seg ordering; near-minimal atomics when clause ids
// repeat over consecutive tokens (as they do here).
// ---------------------------------------------------------------------------
__global__ __launch_bounds__(256)
void seg_accum_kernel(const float* __restrict__ seq,
                      const int*   __restrict__ seg,
                      float*       __restrict__ sums,
                      int n_clauses) {
    __shared__ int s_seg[TOK_PER_BLK];
    const int tid = threadIdx.x;
    const int t0  = blockIdx.x * TOK_PER_BLK;

    if (tid < TOK_PER_BLK) s_seg[tid] = seg[t0 + tid];   // ds_store / ds_load path
    __syncthreads();

    const int col = tid * COLS_PER_THR;                  // 0..765
    float a0 = 0.f, a1 = 0.f, a2 = 0.f;
    int cur = n_clauses;                                 // sentinel: nothing pending

    #pragma unroll 4
    for (int i = 0; i < TOK_PER_BLK; ++i) {
        int s = s_seg[i];                                // block-uniform -> scalar branch
        if (s != cur) {
            if (cur < n_clauses) {
                float* p = sums + (size_t)cur * HID + col;
                atomicAdd(p + 0, a0);
                atomicAdd(p + 1, a1);
                atomicAdd(p + 2, a2);
            }
            a0 = a1 = a2 = 0.f;
            cur = s;
        }
        if (s < n_clauses) {                             // skip unmasked rows entirely
            const float* r = seq + (size_t)(t0 + i) * HID + col;
            a0 += r[0];                                  // merged into b96 load
            a1 += r[1];
            a2 += r[2];
        }
    }
    if (cur < n_clauses) {
        float* p = sums + (size_t)cur * HID + col;
        atomicAdd(p + 0, a0);
        atomicAdd(p + 1, a1);
        atomicAdd(p + 2, a2);
    }
}

// ---------------------------------------------------------------------------
// Kernel 3: logits[NC][16] = (sums / counts) @ fc_w^T + fc_b via
// V_WMMA_F32_16X16X4_F32. One wave per 16-clause tile; K-loop of 192 WMMAs.
//
// A (16x4 f32) layout per ISA: lane m(0..15)/m+16, VGPR0 = K{0,2}, VGPR1 = K{1,3}
//   -> a.x = A[m][k + 2*half], a.y = A[m][k + 2*half + 1]
// B (4x16) mirrors A with M<->N; B[k][n] = fc_w[n][k].
// C/D (16x16 f32): VGPR g, lane l -> M = g + 8*(l>>4), N = l&15.
// EXEC must be all 1s: grid sized so every wave owns a valid tile, no
// divergent control flow in the loop.
// ---------------------------------------------------------------------------
__global__ __launch_bounds__(128)
void wmma_fc_kernel(const float* __restrict__ sums,
                    const float* __restrict__ fc_w,
                    const float* __restrict__ fc_b,
                    const float* __restrict__ counts,
                    float*       __restrict__ out) {
    const int wave = (blockIdx.x * blockDim.x + threadIdx.x) >> 5;  // tile id
    const int lane = threadIdx.x & 31;
    const int half = lane >> 4;       // 0: lanes 0-15, 1: lanes 16-31
    const int mn   = lane & 15;       // A-row (for loads) / D-col (for stores)

    const int row = wave * 16 + mn;                    // clause row this lane feeds
    const float rcnt = 1.0f / counts[row];             // fold mean into A load
    const float* __restrict__ arow = sums + (size_t)row * HID;
    const float* __restrict__ brow = fc_w + (size_t)mn * HID;   // fc_w[n][*]

    v8f c = {};
    #pragma unroll 4
    for (int k = 0; k < HID; k += 4) {
        v2f a, b;
        a.x = arow[k + 2 * half]     * rcnt;
        a.y = arow[k + 2 * half + 1] * rcnt;
        b.x = brow[k + 2 * half];
        b.y = brow[k + 2 * half + 1];
        // 8 args: (neg_a, A, neg_b, B, c_mod, C, reuse_a, reuse_b)
        c = __builtin_amdgcn_wmma_f32_16x16x4_f32(
                false, a, false, b, (short)0, c, false, false);
    }

    const float bias = fc_b[mn];
    float* __restrict__ orow = out + (size_t)(wave * 16) * 16;
    #pragma unroll
    for (int g = 0; g < 8; ++g) {
        int m = g + 8 * half;
        orow[m * 16 + mn] = c[g] + bias;
    }
}

// ---------------------------------------------------------------------------
extern "C" void kernel_launch(void* const* d_in, const int* in_sizes, int n_in,
                              void* d_out, int out_size, void* d_ws, size_t ws_size,
                              hipStream_t stream) {
    const float* seq    = (const float*)d_in[0];   // [64,512,768] f32
    const float* fc_w   = (const float*)d_in[1];   // [16,768]     f32
    const float* fc_b   = (const float*)d_in[2];   // [16]         f32
    const float* counts = (const float*)d_in[3];   // [1536]       f32
    const int*   seg    = (const int*)d_in[4];     // [64,512]     i32
    (void)in_sizes; (void)n_in; (void)out_size; (void)ws_size;

    float* sums = (float*)d_ws;                    // [NC][H] f32 accumulator
    float* out  = (float*)d_out;                   // [NC][16] f32

    // 1) zero accumulator (NC*H floats = 294912 float4s)
    const int n4 = NCLAUSE * HID / 4;
    zero_f4_kernel<<<(n4 + 255) / 256, 256, 0, stream>>>((float4*)d_ws, n4);

    // 2) segment accumulate: 32768/64 = 512 blocks x 256 threads
    seg_accum_kernel<<<NTOK / TOK_PER_BLK, 256, 0, stream>>>(seq, seg, sums, NCLAUSE);

    // 3) WMMA FC head: NC/16 = 96 waves -> 24 blocks x 128 threads (4 waves)
    wmma_fc_kernel<<<(NCLAUSE / 16) / 4, 128, 0, stream>>>(sums, fc_w, fc_b, counts, out);
}